// GCN_463856468221
// MI455X (gfx1250) — compile-verified
//
#include <hip/hip_runtime.h>
#include <hip/hip_bf16.h>

typedef float v2f __attribute__((ext_vector_type(2)));
typedef float v8f __attribute__((ext_vector_type(8)));

static __device__ __forceinline__ int imin(int a, int b) { return a < b ? a : b; }

// ---------------------------------------------------------------------------
// Utility: fill a float buffer with a constant (deg init / agg zeroing).
// ---------------------------------------------------------------------------
__global__ __launch_bounds__(256) void k_fill(float* __restrict__ p, float v, int n) {
    int i = blockIdx.x * blockDim.x + threadIdx.x;
    if (i < n) p[i] = v;
}

// Per-edge in-degree accumulation (self-loop already baked in via fill(1.0)).
__global__ __launch_bounds__(256) void k_deg(const long long* __restrict__ ei,
                                             float* __restrict__ deg, int n_edges) {
    int e = blockIdx.x * blockDim.x + threadIdx.x;
    if (e >= n_edges) return;
    int d = (int)ei[(size_t)n_edges + e];
    atomicAdd(deg + d, 1.0f);
}

// In-place deg -> deg^{-1/2}
__global__ __launch_bounds__(256) void k_dinv(float* __restrict__ deg, int n) {
    int i = blockIdx.x * blockDim.x + threadIdx.x;
    if (i >= n) return;
    float d = deg[i];
    deg[i] = (d > 0.0f) ? rsqrtf(d) : 0.0f;
}

// ---------------------------------------------------------------------------
// Dense GEMM  O[N,fout] = A[N,fin] @ W[fin,fout]  via V_WMMA_F32_16X16X4_F32.
// Block = 8 waves, one tile_n per block. W tile (16 cols of W, zero-padded)
// is staged once into LDS in [K][16] layout; each wave computes one 16x16
// output tile. Hot loop: 1x global_load_b64 (A, v2f) + 2x ds_load_b32 (B)
// + 1x v_wmma, with pure pointer increments (no masks, no 64-bit muls).
// Zero-padding of B makes ragged-K A garbage irrelevant (x * 0 == 0); the
// fin%4 tail (layer 1 only) is one masked iteration with in-bounds A loads.
// EXEC is all-1s at every WMMA (wave-uniform control flow only).
// ---------------------------------------------------------------------------
#define LDSW_COLS 16
__global__ __launch_bounds__(256) void k_gemm_wmma(const float* __restrict__ A,
                                                   const float* __restrict__ W,
                                                   float* __restrict__ O,
                                                   int n_rows, int fin, int fout,
                                                   int tiles_n) {
    __shared__ float ldsW[64 * LDSW_COLS];   // K-major, 16 wide, zero-padded

    int tid     = threadIdx.x;
    int waveId  = tid >> 5;
    int lane    = tid & 31;
    int tile_n  = blockIdx.x % tiles_n;
    int tile_m0 = (blockIdx.x / tiles_n) * 8;

    int finp = (fin + 3) & ~3;               // K rounded up to multiple of 4
    // Cooperative stage of W[:, tile_n*16 : +16] into LDS, zero-padded.
    for (int idx = tid; idx < finp * LDSW_COLS; idx += 256) {
        int k   = idx >> 4;
        int c   = idx & 15;
        int col = tile_n * 16 + c;
        float v = 0.0f;
        if (k < fin && col < fout) v = W[(size_t)k * fout + col];
        ldsW[idx] = v;
    }
    __syncthreads();

    int tile_m = tile_m0 + waveId;
    if (tile_m * 16 >= n_rows) return;       // wave-uniform exit

    int half = lane >> 4;                    // 0: K pair {k0,k0+1}, 1: {k0+2,k0+3}
    int l16  = lane & 15;
    int row  = imin(tile_m * 16 + l16, n_rows - 1);

    const float* ap = A + (size_t)row * fin + 2 * half;     // always 8B aligned
    const float* bp = ldsW + 2 * half * LDSW_COLS + l16;

    v8f c = {0.f, 0.f, 0.f, 0.f, 0.f, 0.f, 0.f, 0.f};

    int fin4 = fin & ~3;
    for (int k0 = 0; k0 < fin4; k0 += 4) {
        v2f a = *(const v2f*)ap;
        v2f b;
        b.x = bp[0];
        b.y = bp[LDSW_COLS];
        c = __builtin_amdgcn_wmma_f32_16x16x4_f32(false, a, false, b,
                                                  (short)0, c, false, false);
        ap += 4;
        bp += 4 * LDSW_COLS;
    }
    if (fin4 < fin) {                        // ragged-K tail (fin=34 only)
        int ka  = fin4 + 2 * half;
        int k_0 = imin(ka,     fin - 1);
        int k_1 = imin(ka + 1, fin - 1);
        float m0 = (ka     < fin) ? 1.0f : 0.0f;
        float m1 = (ka + 1 < fin) ? 1.0f : 0.0f;
        v2f a, b;
        a.x = A[(size_t)row * fin + k_0] * m0;
        a.y = A[(size_t)row * fin + k_1] * m1;
        b.x = bp[0];                          // LDS zero-padded for k >= fin
        b.y = bp[LDSW_COLS];
        c = __builtin_amdgcn_wmma_f32_16x16x4_f32(false, a, false, b,
                                                  (short)0, c, false, false);
    }

    int coln = tile_n * 16 + l16;
    if (coln < fout) {
        int mbase = tile_m * 16 + 8 * half;  // C/D: VGPR v <-> row v + 8*half
#pragma unroll
        for (int v = 0; v < 8; ++v) {
            int m = mbase + v;
            if (m < n_rows) O[(size_t)m * fout + coln] = c[v];
        }
    }
}

// ---------------------------------------------------------------------------
// Edge scatter: agg[dst] += h[src] * dinv[src]*dinv[dst].
// One thread handles 4 features of one edge (float4 gather + 4 f32 atomics,
// resolved in L2 on MI455X). cpe = fout/4 is a power of two -> shift indexing.
// ---------------------------------------------------------------------------
__global__ __launch_bounds__(256) void k_scatter(const long long* __restrict__ ei,
                                                 const float* __restrict__ h,
                                                 const float* __restrict__ dinv,
                                                 float* __restrict__ agg,
                                                 int n_edges, int F, int cpe_shift) {
    long long tid = (long long)blockIdx.x * blockDim.x + threadIdx.x;
    int e  = (int)(tid >> cpe_shift);
    int ch = (int)(tid & ((1 << cpe_shift) - 1));
    if (e >= n_edges) return;
    int s = (int)ei[e];
    int d = (int)ei[(size_t)n_edges + e];
    float nrm = dinv[s] * dinv[d];
    const float4 hv = *(const float4*)(h + (size_t)s * F + ch * 4);
    float* ap = agg + (size_t)d * F + ch * 4;
    atomicAdd(ap + 0, hv.x * nrm);
    atomicAdd(ap + 1, hv.y * nrm);
    atomicAdd(ap + 2, hv.z * nrm);
    atomicAdd(ap + 3, hv.w * nrm);
}

// ---------------------------------------------------------------------------
// Fused: out = tanh(agg + hW*dinv^2 (self-loop) + bias).  out may alias agg.
// ---------------------------------------------------------------------------
__global__ __launch_bounds__(256) void k_bias_tanh(const float* agg,
                                                   const float* __restrict__ hw,
                                                   const float* __restrict__ bias,
                                                   const float* __restrict__ dinv,
                                                   float* out, int n_nodes, int F) {
    int t = blockIdx.x * blockDim.x + threadIdx.x;
    int i = t / F;
    int f = t - i * F;
    if (i >= n_nodes) return;
    float di = dinv[i];
    size_t idx = (size_t)i * F + f;
    float v = agg[idx] + hw[idx] * di * di + bias[f];
    out[idx] = tanhf(v);
}

// Final classifier: out[i,0:2] = h4[i,0:8] @ Wc[8,2] + bc
__global__ __launch_bounds__(256) void k_cls(const float* __restrict__ h,
                                             const float* __restrict__ Wc,
                                             const float* __restrict__ bc,
                                             float* __restrict__ out, int n) {
    int i = blockIdx.x * blockDim.x + threadIdx.x;
    if (i >= n) return;
    float o0 = bc[0], o1 = bc[1];
#pragma unroll
    for (int f = 0; f < 8; ++f) {
        float hv = h[(size_t)i * 8 + f];
        o0 = fmaf(hv, Wc[2 * f + 0], o0);
        o1 = fmaf(hv, Wc[2 * f + 1], o1);
    }
    out[2 * i + 0] = o0;
    out[2 * i + 1] = o1;
}

// ---------------------------------------------------------------------------
extern "C" void kernel_launch(void* const* d_in, const int* in_sizes, int n_in,
                              void* d_out, int out_size, void* d_ws, size_t ws_size,
                              hipStream_t stream) {
    const float*     x  = (const float*)d_in[0];
    const long long* ei = (const long long*)d_in[1];   // int64 edge_index [2,E]
    const float* W1 = (const float*)d_in[2];
    const float* b1 = (const float*)d_in[3];
    const float* W2 = (const float*)d_in[4];
    const float* b2 = (const float*)d_in[5];
    const float* W3 = (const float*)d_in[6];
    const float* b3 = (const float*)d_in[7];
    const float* W4 = (const float*)d_in[8];
    const float* b4 = (const float*)d_in[9];
    const float* Wc = (const float*)d_in[10];
    const float* bc = (const float*)d_in[11];
    float* out = (float*)d_out;

    const int n_nodes = in_sizes[0] / 34;
    const int n_edges = in_sizes[1] / 2;

    float* ws   = (float*)d_ws;
    float* dinv = ws;                                          // n_nodes
    size_t offA = ((size_t)n_nodes + 1023) & ~(size_t)1023;
    float* bufA = ws + offA;                                   // n_nodes*64 max
    float* bufB = bufA + (size_t)n_nodes * 64;                 // n_nodes*64 max

    const int T = 256;
    // 1) degree (self-loop = 1) -> dinv = deg^{-1/2}
    k_fill<<<(n_nodes + T - 1) / T, T, 0, stream>>>(dinv, 1.0f, n_nodes);
    k_deg <<<(n_edges + T - 1) / T, T, 0, stream>>>(ei, dinv, n_edges);
    k_dinv<<<(n_nodes + T - 1) / T, T, 0, stream>>>(dinv, n_nodes);

    auto run_layer = [&](const float* hin, int fin, const float* W, const float* b,
                         float* gemm_out, float* agg, float* hout, int fout) {
        int tiles_m = (n_nodes + 15) / 16;
        int tiles_n = (fout + 15) / 16;
        int blocks  = tiles_n * ((tiles_m + 7) / 8);           // 8 waves / block
        k_gemm_wmma<<<blocks, T, 0, stream>>>(hin, W, gemm_out,
                                              n_nodes, fin, fout, tiles_n);
        int nz = n_nodes * fout;
        k_fill<<<(nz + T - 1) / T, T, 0, stream>>>(agg, 0.0f, nz);
        int cpe_shift = __builtin_ctz(fout / 4);
        long long tot = (long long)n_edges << cpe_shift;
        k_scatter<<<(int)((tot + T - 1) / T), T, 0, stream>>>(ei, gemm_out, dinv, agg,
                                                              n_edges, fout, cpe_shift);
        k_bias_tanh<<<(nz + T - 1) / T, T, 0, stream>>>(agg, gemm_out, b, dinv,
                                                        hout, n_nodes, fout);
    };

    float* h4 = out + (size_t)2 * n_nodes;  // second tuple output: h [N,8]

    run_layer(x,    34, W1, b1, bufA, bufB, bufB, 64);
    run_layer(bufB, 64, W2, b2, bufA, bufB, bufB, 32);
    run_layer(bufB, 32, W3, b3, bufA, bufB, bufB, 16);
    run_layer(bufB, 16, W4, b4, bufA, bufB, h4,   8);

    k_cls<<<(n_nodes + T - 1) / T, T, 0, stream>>>(h4, Wc, bc, out, n_nodes);
}